// NECTAR_Binning_28080496181346
// MI455X (gfx1250) — compile-verified
//
#include <hip/hip_runtime.h>
#include <stdint.h>

#define NBINS 15
#define NCLS  4
#define NW2   9
#define HDIM  512
#define WDIM  512
#define BDIM  16
#define TILE  32
#define HALO  34      // TILE + 2 (1-pixel halo each side)
#define LSTR  36      // padded LDS row stride (elements)

// address-space helper types for the async-copy builtins (int-typed params)
typedef __attribute__((address_space(1))) int __as1_int;
typedef __attribute__((address_space(3))) int __as3_int;

__device__ __forceinline__ void async_f32_to_lds(const float* gp, float* lp) {
#if __has_builtin(__builtin_amdgcn_global_load_async_to_lds_b32)
    __builtin_amdgcn_global_load_async_to_lds_b32((__as1_int*)gp, (__as3_int*)lp, 0, 0);
#else
    *lp = *gp;
#endif
}

__global__ __launch_bounds__(256)
void NECTAR_Binning_calib_kernel(const float* __restrict__ logits,
                                 const float* __restrict__ vfreq,
                                 float* __restrict__ out)
{
    __shared__ float    s_log[NCLS * HALO * LSTR];   // 4*34*36 floats = 19584 B
    __shared__ float    s_vf[NCLS * NW2 * NBINS];    // 540 floats
    __shared__ unsigned s_w[HALO * LSTR];            // packed class-weight words, 4896 B

    const int tid = threadIdx.x;
    const int bw0 = blockIdx.x * TILE;
    const int bh0 = blockIdx.y * TILE;
    const int b   = blockIdx.z;
    const long planeHW = (long)HDIM * WDIM;
    const bool interior = (bw0 > 0) & (bw0 < WDIM - TILE) & (bh0 > 0) & (bh0 < HDIM - TILE);

    // ---- Phase 1a: async copy calibration table -> LDS ----
    for (int i = tid; i < NCLS * NW2 * NBINS; i += 256)
        async_f32_to_lds(vfreq + i, &s_vf[i]);

    // ---- Phase 1b: async copy the 4-plane haloed logits tile -> LDS ----
    const float* lbase = logits + (long)b * NCLS * planeHW;
    if (interior) {
        // no clamping needed: whole halo is in-image
        for (int i = tid; i < NCLS * HALO * HALO; i += 256) {
            int ch  = i / (HALO * HALO);
            int rem = i - ch * (HALO * HALO);
            int r   = rem / HALO;
            int c   = rem - r * HALO;
            const float* gp = lbase + (long)ch * planeHW
                            + (long)(bh0 - 1 + r) * WDIM + (bw0 - 1 + c);
            async_f32_to_lds(gp, &s_log[ch * (HALO * LSTR) + r * LSTR + c]);
        }
    } else {
        for (int i = tid; i < NCLS * HALO * HALO; i += 256) {
            int ch  = i / (HALO * HALO);
            int rem = i - ch * (HALO * HALO);
            int r   = rem / HALO;
            int c   = rem - r * HALO;
            int gh  = bh0 - 1 + r; gh = gh < 0 ? 0 : (gh > HDIM - 1 ? HDIM - 1 : gh);
            int gw  = bw0 - 1 + c; gw = gw < 0 ? 0 : (gw > WDIM - 1 ? WDIM - 1 : gw);
            const float* gp = lbase + (long)ch * planeHW + (long)gh * WDIM + gw;
            async_f32_to_lds(gp, &s_log[ch * (HALO * LSTR) + r * LSTR + c]);
        }
    }
#if __has_builtin(__builtin_amdgcn_global_load_async_to_lds_b32)
#if __has_builtin(__builtin_amdgcn_s_wait_asynccnt)
    __builtin_amdgcn_s_wait_asynccnt(0);
#else
    asm volatile("s_wait_asynccnt 0" ::: "memory");
#endif
#endif
    __syncthreads();

    // ---- Phase 2: argmax class -> packed weight word (byte lane per class) ----
    // w = 1 << (8*argmax) for in-image positions, 0 outside (matches zero-padded onehot)
    for (int i = tid; i < HALO * HALO; i += 256) {
        int r = i / HALO;
        int c = i - r * HALO;
        float best = s_log[r * LSTR + c];
        int   bi   = 0;
#pragma unroll
        for (int ch = 1; ch < NCLS; ++ch) {
            float v = s_log[ch * (HALO * LSTR) + r * LSTR + c];
            if (v > best) { best = v; bi = ch; }   // strict '>' = first-max (jnp.argmax)
        }
        unsigned w = 1u << (bi * 8);
        if (!interior) {
            int gh = bh0 - 1 + r;
            int gw = bw0 - 1 + c;
            if (gh < 0 || gh >= HDIM || gw < 0 || gw >= WDIM) w = 0u;
        }
        s_w[r * LSTR + c] = w;
    }
    __syncthreads();

    // ---- Phase 3: softmax + bins + packed neighbor counts + gather + normalize ----
    const int tx = tid & 31;
    const int ty = tid >> 5;  // 0..7

#pragma unroll
    for (int k = 0; k < 4; ++k) {
        const int row = ty + k * 8;   // 0..31 within tile
        const int r   = row + 1;      // halo coords
        const int c   = tx + 1;

        float l[NCLS];
#pragma unroll
        for (int ch = 0; ch < NCLS; ++ch)
            l[ch] = s_log[ch * (HALO * LSTR) + r * LSTR + c];

        float m = fmaxf(fmaxf(l[0], l[1]), fmaxf(l[2], l[3]));
        float e[NCLS];
        float ssum = 0.f;
#pragma unroll
        for (int ch = 0; ch < NCLS; ++ch) { e[ch] = __expf(l[ch] - m); ssum += e[ch]; }
        const float inv = 1.0f / ssum;   // IEEE divide: feeds discrete bin selection

        // 3x3 packed sum minus center: byte lane ch = count of neighbors with class ch
        unsigned acc = 0u;
#pragma unroll
        for (int dr = -1; dr <= 1; ++dr)
#pragma unroll
            for (int dc = -1; dc <= 1; ++dc)
                acc += s_w[(r + dr) * LSTR + (c + dc)];
        acc -= s_w[r * LSTR + c];        // exclude center (kernel center = 0)

        float v[NCLS];
        float tot = 0.f;
#pragma unroll
        for (int ch = 0; ch < NCLS; ++ch) {
            float p   = e[ch] * inv;                  // p in [0,1]
            int   bin = (int)(p * (float)NBINS);      // trunc == floor for p >= 0
            bin = bin > (NBINS - 1) ? (NBINS - 1) : bin;
            int   cnt = (int)((acc >> (ch * 8)) & 0xFFu);   // <= 8, no clip needed
            v[ch] = s_vf[ch * (NW2 * NBINS) + cnt * NBINS + bin];
            tot  += v[ch];
        }
        if (tot == 0.0f) tot = 1e-8f;
        // final normalize is continuous in tot -> 1-ulp hardware rcp is safe
#if __has_builtin(__builtin_amdgcn_rcpf)
        const float invt = __builtin_amdgcn_rcpf(tot);
#else
        const float invt = 1.0f / tot;
#endif

        const long outbase = (long)b * NCLS * planeHW + (long)(bh0 + row) * WDIM + (bw0 + tx);
#pragma unroll
        for (int ch = 0; ch < NCLS; ++ch)
            out[outbase + ch * planeHW] = v[ch] * invt;
    }
}

extern "C" void kernel_launch(void* const* d_in, const int* in_sizes, int n_in,
                              void* d_out, int out_size, void* d_ws, size_t ws_size,
                              hipStream_t stream) {
    (void)in_sizes; (void)n_in; (void)out_size; (void)d_ws; (void)ws_size;
    const float* logits = (const float*)d_in[0];   // [16,4,512,512] f32
    const float* vfreq  = (const float*)d_in[1];   // [4,9,15] f32
    float*       out    = (float*)d_out;           // [16,4,512,512] f32

    dim3 grid(WDIM / TILE, HDIM / TILE, BDIM);     // 16 x 16 x 16 tiles
    NECTAR_Binning_calib_kernel<<<grid, 256, 0, stream>>>(logits, vfreq, out);
}